// SelfDistillationGraphConvGNN_42528766165144
// MI455X (gfx1250) — compile-verified
//
#include <hip/hip_runtime.h>
#include <hip/hip_bf16.h>

// ---------------------------------------------------------------------------
// SelfDistillationGraphConvGNN on MI455X (gfx1250), fp32 end-to-end.
// GEMMs use V_WMMA_F32_16X16X4_F32 (exact fp32, wave32 WMMA).
// ---------------------------------------------------------------------------

typedef float v2f __attribute__((ext_vector_type(2)));
typedef float v8f __attribute__((ext_vector_type(8)));

#define NV      100000
#define NS      200000
#define HID     128
#define BATCH   1024
#define NCLS    4
#define NSLOT   64          // stats accumulation slots (atomic decongestion)
#define EPSBN   1e-5f

// ------------------------------ helpers ------------------------------------

__device__ __forceinline__ unsigned f2key(float f) {
    unsigned u = __float_as_uint(f);
    return (u & 0x80000000u) ? ~u : (u | 0x80000000u);   // order-preserving map
}
__device__ __forceinline__ float key2f(unsigned k) {
    unsigned u = (k & 0x80000000u) ? (k & 0x7FFFFFFFu) : ~k;
    return __uint_as_float(u);
}

__global__ void fill_zero_k(float* __restrict__ p, long n) {
    long i = (long)blockIdx.x * blockDim.x + threadIdx.x;
    if (i < n) p[i] = 0.0f;
}

__global__ void count_k(const int* __restrict__ batch, float* __restrict__ cnt, int n) {
    int i = blockIdx.x * blockDim.x + threadIdx.x;
    if (i < n) atomicAdd(&cnt[batch[i]], 1.0f);
}

// ------------------------- edge gather/scatter ------------------------------
// One thread handles one float4 chunk of one edge. agg has fixed stride HID.

__global__ void scatter_add_k(const float* __restrict__ xsrc, int srcStride,
                              const int* __restrict__ edges, int E,
                              float* __restrict__ agg, int cpeShift) {
    long gid   = (long)blockIdx.x * blockDim.x + threadIdx.x;
    long total = (long)E << cpeShift;
    if (gid >= total) return;
    int e  = (int)(gid >> cpeShift);
    int ch = ((int)gid & ((1 << cpeShift) - 1)) << 2;       // float offset
    int src = edges[e];
    int dst = edges[E + e];
    float4 v = *(const float4*)(xsrc + (long)src * srcStride + ch);
    float* d = agg + (long)dst * HID + ch;
    atomicAdd(d + 0, v.x);
    atomicAdd(d + 1, v.y);
    atomicAdd(d + 2, v.z);
    atomicAdd(d + 3, v.w);
}

// ------------------------------ fused GEMM ----------------------------------
// out[rows x 128] = relu( A@Wrel + X@Wroot + brel ), in place over A
// (A stride is always HID; only cols [0,K) are read).
// Block = 16 rows x 128 cols, 8 waves; each wave owns a 16-col tile and chains
// v_wmma_f32_16x16x4_f32 over K. Also emits per-channel sum/sumsq partials.
//
// Frag layouts (ISA 7.12.2, 32-bit):
//   A 16x4 : lane<16 -> M=lane, K={k0,k0+1}; lane>=16 -> M=lane-16, K={k0+2,k0+3}
//   B 4x16 : lane<16 -> N=lane, K={k0,k0+1}; lane>=16 -> N=lane-16, K={k0+2,k0+3}
//   C 16x16: vgpr j, lane<16 -> M=j, N=lane; lane>=16 -> M=j+8, N=lane-16

template <int K>
__global__ __launch_bounds__(256) void gemm_fused_k(
        const float* __restrict__ A,
        const float* __restrict__ X, int ldX,
        const float* __restrict__ Wrel, const float* __restrict__ Wroot,
        const float* __restrict__ brel,
        float* __restrict__ out,               // == A (in place, stride HID)
        float* __restrict__ stats)             // [NSLOT][2][HID]
{
    const int w      = threadIdx.x >> 5;
    const int lane   = threadIdx.x & 31;
    const int lane15 = lane & 15;
    const int khalf  = lane >> 4;              // 0: K={0,1}+k0, 1: K={2,3}+k0
    const int row0   = blockIdx.x << 4;
    const int col    = (w << 4) + lane15;

    const float* arow = A + (long)(row0 + lane15) * HID;
    const float* xrow = X + (long)(row0 + lane15) * ldX;

    v8f c = {0.f, 0.f, 0.f, 0.f, 0.f, 0.f, 0.f, 0.f};

#pragma unroll
    for (int k0 = 0; k0 < K; k0 += 4) {
        int ka = k0 + khalf * 2;
        v2f a; a.x = arow[ka];            a.y = arow[ka + 1];
        v2f b; b.x = Wrel[ka * HID + col]; b.y = Wrel[(ka + 1) * HID + col];
        c = __builtin_amdgcn_wmma_f32_16x16x4_f32(false, a, false, b,
                                                  (short)0, c, false, false);
    }
#pragma unroll
    for (int k0 = 0; k0 < K; k0 += 4) {
        int ka = k0 + khalf * 2;
        v2f a; a.x = xrow[ka];             a.y = xrow[ka + 1];
        v2f b; b.x = Wroot[ka * HID + col]; b.y = Wroot[(ka + 1) * HID + col];
        c = __builtin_amdgcn_wmma_f32_16x16x4_f32(false, a, false, b,
                                                  (short)0, c, false, false);
    }

    __syncthreads();   // all in-block reads of A done before in-place writes

    float bias = brel[col];
    float s = 0.f, sq = 0.f;
#pragma unroll
    for (int j = 0; j < 8; ++j) {
        float v = c[j] + bias;
        v = v > 0.f ? v : 0.f;                       // relu
        int row = row0 + j + khalf * 8;
        out[(long)row * HID + col] = v;
        s  += v;
        sq += v * v;
    }
    // combine lane pairs sharing the same column (lane ^ 16)
    s  += __shfl_xor(s, 16);
    sq += __shfl_xor(sq, 16);
    if (khalf == 0) {
        float* st = stats + (blockIdx.x & (NSLOT - 1)) * (2 * HID);
        atomicAdd(st + col, s);
        atomicAdd(st + HID + col, sq);
    }
}

// --------------------------- batchnorm finalize -----------------------------

__global__ void bn_finalize_k(const float* __restrict__ stats,
                              const float* __restrict__ gamma,
                              const float* __restrict__ beta,
                              float* __restrict__ bnp, float n) {
    int c = threadIdx.x;                      // 128 threads
    float s = 0.f, sq = 0.f;
    for (int k = 0; k < NSLOT; ++k) {
        s  += stats[k * 2 * HID + c];
        sq += stats[k * 2 * HID + HID + c];
    }
    float mu  = s / n;
    float var = sq / n - mu * mu;
    var = var < 0.f ? 0.f : var;
    float inv   = rsqrtf(var + EPSBN);
    float scale = gamma[c] * inv;
    bnp[c]       = scale;
    bnp[HID + c] = beta[c] - mu * scale;
}

// --------------------- batchnorm apply + segment pooling --------------------

__global__ void bn_apply_pool_k(const float* __restrict__ newbuf,
                                const float* __restrict__ bnp,
                                const int* __restrict__ batch,
                                float* __restrict__ xout,
                                float* __restrict__ psum,
                                unsigned* __restrict__ pmax, int nrows) {
    long gid   = (long)blockIdx.x * blockDim.x + threadIdx.x;
    long total = (long)nrows * 32;            // 32 threads/node, 4 ch each
    if (gid >= total) return;
    int node = (int)(gid >> 5);
    int c0   = ((int)gid & 31) << 2;
    int b    = batch[node];

    float4 v  = *(const float4*)(newbuf + (long)node * HID + c0);
    float4 sc = *(const float4*)(bnp + c0);
    float4 sh = *(const float4*)(bnp + HID + c0);
    float4 r;
    r.x = v.x * sc.x + sh.x;
    r.y = v.y * sc.y + sh.y;
    r.z = v.z * sc.z + sh.z;
    r.w = v.w * sc.w + sh.w;
    *(float4*)(xout + (long)node * HID + c0) = r;

    float* ps = psum + (long)b * HID + c0;
    atomicAdd(ps + 0, r.x);
    atomicAdd(ps + 1, r.y);
    atomicAdd(ps + 2, r.z);
    atomicAdd(ps + 3, r.w);
    unsigned* pm = pmax + (long)b * HID + c0;
    atomicMax(pm + 0, f2key(r.x));
    atomicMax(pm + 1, f2key(r.y));
    atomicMax(pm + 2, f2key(r.z));
    atomicMax(pm + 3, f2key(r.w));
}

// ------------------------------ pool finalize -------------------------------

__global__ void pool_finalize_k(const float* __restrict__ vsum,
                                const float* __restrict__ ssum,
                                const unsigned* __restrict__ vmax,
                                const unsigned* __restrict__ smax,
                                const float* __restrict__ vcnt,
                                const float* __restrict__ scnt,
                                float* __restrict__ readout,
                                float* __restrict__ feat_out) {
    int gid = blockIdx.x * blockDim.x + threadIdx.x;   // BATCH*HID
    if (gid >= BATCH * HID) return;
    int b = gid >> 7, c = gid & 127;
    float cv = vcnt[b], cs = scnt[b];
    float mean = vsum[gid] / fmaxf(cv, 1.f) + ssum[gid] / fmaxf(cs, 1.f);
    float mx   = (cv > 0.f ? key2f(vmax[gid]) : 0.f)
               + (cs > 0.f ? key2f(smax[gid]) : 0.f);
    long base = (long)b * (2 * HID);
    float r0 = readout[base + c] + mean;
    float r1 = readout[base + HID + c] + mx;
    readout[base + c]        = r0;
    readout[base + HID + c]  = r1;
    feat_out[base + c]       = r0;
    feat_out[base + HID + c] = r1;
}

// --------------------------------- FC head ----------------------------------

__global__ void fc_k(const float* __restrict__ feat, const float* __restrict__ W,
                     const float* __restrict__ bias, float* __restrict__ out) {
    int gid = blockIdx.x * blockDim.x + threadIdx.x;   // BATCH*NCLS
    if (gid >= BATCH * NCLS) return;
    int b = gid >> 2, j = gid & 3;
    float acc = bias[j];
    const float* f = feat + (long)b * (2 * HID);
    for (int c = 0; c < 2 * HID; ++c) acc += f[c] * W[c * NCLS + j];
    out[gid] = acc;
}

// ------------------------------- host driver --------------------------------

static inline unsigned gblocks(long n, int b) { return (unsigned)((n + b - 1) / b); }

extern "C" void kernel_launch(void* const* d_in, const int* in_sizes, int n_in,
                              void* d_out, int out_size, void* d_ws, size_t ws_size,
                              hipStream_t stream) {
    (void)n_in; (void)out_size; (void)ws_size;

    // --- inputs (setup_inputs dict order) ---
    const float* x_visit   = (const float*)d_in[0];   // [NV,64]
    const float* x_service = (const float*)d_in[1];   // [NS,64]
    const int*   e_vs      = (const int*)d_in[2];     // [2,E]
    const int*   e_sv      = (const int*)d_in[3];
    const int*   b_visit   = (const int*)d_in[4];
    const int*   b_service = (const int*)d_in[5];
    const int    E         = in_sizes[2] / 2;

    // params pytree flattened in JAX tree order (dict keys sorted):
    //  6..9   bn: service.beta, service.gamma, visit.beta, visit.gamma
    //  10..15 fcs[l]: W, b
    //  16+6l  layers[l]: sv.W_rel, sv.W_root, sv.b_rel, vs.W_rel, vs.W_root, vs.b_rel
    const float* bn_s_beta  = (const float*)d_in[6];
    const float* bn_s_gamma = (const float*)d_in[7];
    const float* bn_v_beta  = (const float*)d_in[8];
    const float* bn_v_gamma = (const float*)d_in[9];

    // --- workspace layout (floats) ---
    float* ws = (float*)d_ws;
    float* xv      = ws; ws += (long)NV * HID;
    float* xs      = ws; ws += (long)NS * HID;
    float* readout = ws; ws += (long)BATCH * 2 * HID;
    float* vcnt    = ws; ws += BATCH;
    float* scnt    = ws; ws += BATCH;
    float* zbase   = ws;                       // per-layer zero region starts here
    float* agg_v   = ws; ws += (long)NV * HID;
    float* agg_s   = ws; ws += (long)NS * HID;
    float* stats_v = ws; ws += NSLOT * 2 * HID;
    float* stats_s = ws; ws += NSLOT * 2 * HID;
    float* bnp_v   = ws; ws += 2 * HID;
    float* bnp_s   = ws; ws += 2 * HID;
    float* vsum    = ws; ws += (long)BATCH * HID;
    float* ssum    = ws; ws += (long)BATCH * HID;
    unsigned* vmaxk = (unsigned*)ws; ws += (long)BATCH * HID;
    unsigned* smaxk = (unsigned*)ws; ws += (long)BATCH * HID;
    const long zcount = ws - zbase;

    float* out_base  = (float*)d_out;                       // 3 x [BATCH,NCLS]
    float* feat_base = out_base + 3L * BATCH * NCLS;        // 3 x [BATCH,2H]

    // --- one-time init: readout + counts ---
    {
        long n = (long)BATCH * 2 * HID + 2 * BATCH;         // readout,vcnt,scnt contiguous
        fill_zero_k<<<gblocks(n, 256), 256, 0, stream>>>(readout, n);
        count_k<<<gblocks(NV, 256), 256, 0, stream>>>(b_visit, vcnt, NV);
        count_k<<<gblocks(NS, 256), 256, 0, stream>>>(b_service, scnt, NS);
    }

    const float* xv_cur = x_visit;   int ldv = 64;
    const float* xs_cur = x_service; int lds = 64;

    for (int l = 0; l < 3; ++l) {
        const int K        = (l == 0) ? 64 : 128;
        const int cpeShift = (l == 0) ? 4 : 5;              // K/4 float4 chunks
        const float* sv_Wrel  = (const float*)d_in[16 + 6 * l + 0];
        const float* sv_Wroot = (const float*)d_in[16 + 6 * l + 1];
        const float* sv_b     = (const float*)d_in[16 + 6 * l + 2];
        const float* vs_Wrel  = (const float*)d_in[16 + 6 * l + 3];
        const float* vs_Wroot = (const float*)d_in[16 + 6 * l + 4];
        const float* vs_b     = (const float*)d_in[16 + 6 * l + 5];
        const float* fc_W     = (const float*)d_in[10 + 2 * l];
        const float* fc_b     = (const float*)d_in[11 + 2 * l];

        // zero agg/stats/pool region
        fill_zero_k<<<gblocks(zcount, 256), 256, 0, stream>>>(zbase, zcount);

        // edge scatter: vs (visit -> service agg), sv (service -> visit agg)
        scatter_add_k<<<gblocks((long)E << cpeShift, 256), 256, 0, stream>>>(
            xv_cur, ldv, e_vs, E, agg_s, cpeShift);
        scatter_add_k<<<gblocks((long)E << cpeShift, 256), 256, 0, stream>>>(
            xs_cur, lds, e_sv, E, agg_v, cpeShift);

        // fused GEMM + bias + relu + BN stats (in place over agg)
        if (K == 64) {
            gemm_fused_k<64><<<NS / 16, 256, 0, stream>>>(
                agg_s, xs_cur, lds, vs_Wrel, vs_Wroot, vs_b, agg_s, stats_s);
            gemm_fused_k<64><<<NV / 16, 256, 0, stream>>>(
                agg_v, xv_cur, ldv, sv_Wrel, sv_Wroot, sv_b, agg_v, stats_v);
        } else {
            gemm_fused_k<128><<<NS / 16, 256, 0, stream>>>(
                agg_s, xs_cur, lds, vs_Wrel, vs_Wroot, vs_b, agg_s, stats_s);
            gemm_fused_k<128><<<NV / 16, 256, 0, stream>>>(
                agg_v, xv_cur, ldv, sv_Wrel, sv_Wroot, sv_b, agg_v, stats_v);
        }

        // BN finalize (fold gamma/beta into scale/shift)
        bn_finalize_k<<<1, HID, 0, stream>>>(stats_s, bn_s_gamma, bn_s_beta, bnp_s, (float)NS);
        bn_finalize_k<<<1, HID, 0, stream>>>(stats_v, bn_v_gamma, bn_v_beta, bnp_v, (float)NV);

        // BN apply + segment pooling, writes new xv/xs
        bn_apply_pool_k<<<gblocks((long)NS * 32, 256), 256, 0, stream>>>(
            agg_s, bnp_s, b_service, xs, ssum, smaxk, NS);
        bn_apply_pool_k<<<gblocks((long)NV * 32, 256), 256, 0, stream>>>(
            agg_v, bnp_v, b_visit, xv, vsum, vmaxk, NV);

        // readout update + feature_list[l] + output_list[l]
        float* feat_l = feat_base + (long)l * BATCH * 2 * HID;
        float* out_l  = out_base + (long)l * BATCH * NCLS;
        pool_finalize_k<<<gblocks((long)BATCH * HID, 256), 256, 0, stream>>>(
            vsum, ssum, vmaxk, smaxk, vcnt, scnt, readout, feat_l);
        fc_k<<<gblocks((long)BATCH * NCLS, 256), 256, 0, stream>>>(
            feat_l, fc_W, fc_b, out_l);

        xv_cur = xv; ldv = HID;
        xs_cur = xs; lds = HID;
    }
}